// MemorySystem_56444460204436
// MI455X (gfx1250) — compile-verified
//
#include <hip/hip_runtime.h>
#include <hip/hip_bf16.h>

// MI455X / gfx1250, wave32. fp32 V_WMMA_F32_16X16X4_F32 used for both matvecs.
// Problem sizes are fixed by the reference: E=4096, M=65536, K3=3*E=12288, TOPK=32.

#define E_DIM   4096
#define M_KEYS  65536
#define K3_DIM  12288
#define TOPK    32

typedef __attribute__((ext_vector_type(2))) float v2f;
typedef __attribute__((ext_vector_type(8))) float v8f;

// ---------------------------------------------------------------------------
// Kernel 1: sims[m] = dot(keys[m], q) / max(||keys[m]||, eps)
// (q_norm is a positive constant scale -> does not affect top-k ordering)
//
// WMMA mapping per wave-tile of 16 rows:
//   A (16x4 f32): all 16 rows = q[k..k+3]   -> per-lane a = {q[k+hi*2], q[k+hi*2+1]}
//   B (4x16 f32): column n = keys row (mb+n) -> per-lane b = {krow[k+hi*2], krow[k+hi*2+1]}
//   where hi = lane>=16, row = mb + (lane&15)   (ISA 16x4 f32 layout: K=(lane/16)*2+vgpr)
//   C[m][n] = dot(keys[mb+n], q) for every m -> c[0] on lanes 0..15 holds the 16 dots.
// Row norms accumulate on the VALU alongside (matrix pipe does the dots).
// ---------------------------------------------------------------------------
__global__ __launch_bounds__(256) void sims_wmma_kernel(
    const float* __restrict__ keys, const float* __restrict__ q,
    float* __restrict__ sims)
{
    __shared__ float qs[E_DIM];                       // 16 KB
    for (int i = threadIdx.x; i < E_DIM; i += 256) qs[i] = q[i];
    __syncthreads();

    const int lane = threadIdx.x & 31;
    const int wave = threadIdx.x >> 5;
    const int mb   = blockIdx.x * 128 + wave * 16;    // 8 waves * 16 rows / block
    const int hi   = lane >> 4;                       // 0: K%4 in {0,1}, 1: {2,3}
    const int row  = mb + (lane & 15);
    const float* __restrict__ krow = keys + (size_t)row * E_DIM;

    v8f  c    = {};
    float nacc = 0.0f;

#pragma unroll 8
    for (int k = 0; k < E_DIM; k += 4) {
        const float2 av = *reinterpret_cast<const float2*>(qs   + k + hi * 2);
        const float2 bv = *reinterpret_cast<const float2*>(krow + k + hi * 2);
        v2f a; a.x = av.x; a.y = av.y;
        v2f b; b.x = bv.x; b.y = bv.y;
        nacc = fmaf(bv.x, bv.x, fmaf(bv.y, bv.y, nacc));
        c = __builtin_amdgcn_wmma_f32_16x16x4_f32(
                /*neg_a=*/false, a, /*neg_b=*/false, b,
                /*c_mod=*/(short)0, c, /*reuse_a=*/false, /*reuse_b=*/false);
    }

    // lanes L and L+16 each hold half of row (lane&15)'s squared norm
    const float norm2 = nacc + __shfl_xor(nacc, 16, 32);
    if (lane < 16) {
        const float dot = c[0];                        // C[0][n] = dot(row mb+n, q)
        sims[row] = dot / fmaxf(sqrtf(norm2), 1e-8f);
    }
}

// ---------------------------------------------------------------------------
// Kernel 2: iterative top-32 argmax over sims[M]; single workgroup, 1024 thr.
// Tie-break on lower index (matches jax.lax.top_k stability). Destroys sims
// (set picked to -FLT_MAX) -- sims is fully rewritten by kernel 1 every call.
// ---------------------------------------------------------------------------
__global__ __launch_bounds__(1024) void topk_kernel(
    float* __restrict__ sims, int* __restrict__ topk)
{
    __shared__ float sv[1024];
    __shared__ int   si[1024];
    const int tid = threadIdx.x;

    for (int t = 0; t < TOPK; ++t) {
        float best = -__FLT_MAX__;
        int   bi   = 0x7fffffff;
        for (int i = tid; i < M_KEYS; i += 1024) {
            const float v = sims[i];
            if (v > best || (v == best && i < bi)) { best = v; bi = i; }
        }
        sv[tid] = best; si[tid] = bi;
        __syncthreads();
        for (int s = 512; s > 0; s >>= 1) {
            if (tid < s) {
                const float v2 = sv[tid + s]; const int i2 = si[tid + s];
                if (v2 > sv[tid] || (v2 == sv[tid] && i2 < si[tid])) {
                    sv[tid] = v2; si[tid] = i2;
                }
            }
            __syncthreads();
        }
        if (tid == 0) {
            topk[t] = si[0];
            sims[si[0]] = -__FLT_MAX__;   // exclude from later passes
        }
        __syncthreads();
    }
}

// ---------------------------------------------------------------------------
// Kernel 3: combined[0:E]=core, [E:2E]=study, [2E:3E]=mean of 32 gathered rows
// ---------------------------------------------------------------------------
__global__ __launch_bounds__(256) void combine_kernel(
    const float* __restrict__ core, const float* __restrict__ study,
    const float* __restrict__ values, const int* __restrict__ topk,
    float* __restrict__ combined)
{
    __shared__ int tk[TOPK];
    if (threadIdx.x < TOPK) tk[threadIdx.x] = topk[threadIdx.x];
    __syncthreads();

    const int idx = blockIdx.x * 256 + threadIdx.x;   // 48 blocks * 256 = 12288
    if (idx < E_DIM) {
        combined[idx] = core[idx];
    } else if (idx < 2 * E_DIM) {
        combined[idx] = study[idx - E_DIM];
    } else {
        const int e = idx - 2 * E_DIM;
        float s = 0.0f;
#pragma unroll
        for (int j = 0; j < TOPK; ++j)
            s += values[(size_t)tk[j] * E_DIM + e];
        combined[idx] = s * (1.0f / (float)TOPK);
    }
}

// ---------------------------------------------------------------------------
// Kernel 4: out[i] = fusion_b[i] + dot(W_row_i, combined), W: 4096 x 12288.
// Same WMMA wave-tile as kernel 1; combined staged in LDS (48 KB).
// ---------------------------------------------------------------------------
__global__ __launch_bounds__(256) void fusion_wmma_kernel(
    const float* __restrict__ W, const float* __restrict__ bias,
    const float* __restrict__ combined, float* __restrict__ out)
{
    __shared__ float cs[K3_DIM];                      // 48 KB
    for (int i = threadIdx.x; i < K3_DIM; i += 256) cs[i] = combined[i];
    __syncthreads();

    const int lane = threadIdx.x & 31;
    const int wave = threadIdx.x >> 5;
    const int mb   = blockIdx.x * 128 + wave * 16;    // 32 blocks cover 4096 rows
    const int hi   = lane >> 4;
    const int row  = mb + (lane & 15);
    const float* __restrict__ wrow = W + (size_t)row * K3_DIM;

    v8f c = {};
#pragma unroll 8
    for (int k = 0; k < K3_DIM; k += 4) {
        const float2 av = *reinterpret_cast<const float2*>(cs   + k + hi * 2);
        const float2 bv = *reinterpret_cast<const float2*>(wrow + k + hi * 2);
        v2f a; a.x = av.x; a.y = av.y;
        v2f b; b.x = bv.x; b.y = bv.y;
        c = __builtin_amdgcn_wmma_f32_16x16x4_f32(
                false, a, false, b, (short)0, c, false, false);
    }

    if (lane < 16) out[row] = c[0] + bias[row];
}

// ---------------------------------------------------------------------------
// Launch. Workspace layout (needs ~311 KB):
//   [0, 256K)         sims      : 65536 f32
//   [256K, 256K+128)  topk      : 32 i32
//   [256K+256, +48K)  combined  : 12288 f32
// ---------------------------------------------------------------------------
extern "C" void kernel_launch(void* const* d_in, const int* in_sizes, int n_in,
                              void* d_out, int out_size, void* d_ws, size_t ws_size,
                              hipStream_t stream)
{
    const float* core   = (const float*)d_in[0];
    const float* study  = (const float*)d_in[1];
    const float* query  = (const float*)d_in[2];
    const float* keys   = (const float*)d_in[3];
    const float* values = (const float*)d_in[4];
    const float* fw     = (const float*)d_in[5];
    const float* fb     = (const float*)d_in[6];
    // d_in[7] = top_k scalar (32), hardcoded as TOPK

    float* out      = (float*)d_out;
    char*  ws       = (char*)d_ws;
    float* sims     = (float*)(ws);
    int*   topk     = (int*)  (ws + (size_t)M_KEYS * 4);
    float* combined = (float*)(ws + (size_t)M_KEYS * 4 + 256);

    // 1) sims: 65536 rows, 128 rows/block (8 waves x 16) -> 512 blocks
    sims_wmma_kernel<<<512, 256, 0, stream>>>(keys, query, sims);
    // 2) top-32 selection, one workgroup
    topk_kernel<<<1, 1024, 0, stream>>>(sims, topk);
    // 3) build combined vector (12288 elems)
    combine_kernel<<<48, 256, 0, stream>>>(core, study, values, topk, combined);
    // 4) fusion matvec + bias: 4096 rows -> 32 blocks
    fusion_wmma_kernel<<<32, 256, 0, stream>>>(fw, fb, combined, out);
}